// DirGATConv_62723702391595
// MI455X (gfx1250) — compile-verified
//
#include <hip/hip_runtime.h>
#include <hip/hip_bf16.h>

// ---------------------------------------------------------------------------
// DirGATConv for MI455X (gfx1250, wave32).
//  - fp32 GEMM on the matrix pipe via V_WMMA_F32_16X16X4_F32 (exact fp32).
//  - segment softmax with float atomic-max / atomic-add (L2-resident).
//  - wave-per-edge aggregation, direction weights folded into the scatter.
// Workspace usage: ~123 MB of d_ws (h1,h2 = 102.4 MB; logits/max/den = 6.4 MB;
// per-edge softmax numerators p1,p2 = 14.4 MB).
// ---------------------------------------------------------------------------

#define NNODES 50000
#define NEDGES 400000
#define ETOT   (NEDGES + NNODES)   // GATConv adds self-loops
#define DIN    256
#define HEADS  4
#define CH     64
#define HC     (HEADS * CH)        // 256
#define ALPHA_W 0.5f
#define SLOPE   0.2f

typedef __attribute__((ext_vector_type(2))) float v2f;
typedef __attribute__((ext_vector_type(8))) float v8f;

__device__ __forceinline__ float lrelu(float x) { return x > 0.0f ? x : SLOPE * x; }

// float atomic max via signed/unsigned integer monotonicity trick.
// Valid when the location is initialized to -inf.
__device__ __forceinline__ void atomicMaxF(float* addr, float v) {
    if (v >= 0.0f) atomicMax((int*)addr, __float_as_int(v));
    else           atomicMin((unsigned int*)addr, __float_as_uint(v));
}

// ---------------------------------------------------------------------------
// Phase 0: init. out <- (1-a)*b1 + a*b2 ; emax <- -inf ; den <- 0.
// ---------------------------------------------------------------------------
__global__ void __launch_bounds__(256)
init_kernel(float* __restrict__ out, const float* __restrict__ b1,
            const float* __restrict__ b2, float* __restrict__ emax1,
            float* __restrict__ emax2, float* __restrict__ den1,
            float* __restrict__ den2) {
    int i = blockIdx.x * blockDim.x + threadIdx.x;
    if (i < NNODES * HC) {
        int c = i & (HC - 1);
        out[i] = (1.0f - ALPHA_W) * b1[c] + ALPHA_W * b2[c];
    }
    if (i < NNODES * HEADS) {
        emax1[i] = -__builtin_inff();
        emax2[i] = -__builtin_inff();
        den1[i]  = 0.0f;
        den2[i]  = 0.0f;
    }
}

// ---------------------------------------------------------------------------
// Phase 1: H = X @ W  via fp32 WMMA.  One block = 16 rows x 256 cols.
// 4 waves/block; each wave owns a 16x64 strip -> 4 v8f accumulators,
// one shared A fragment per K-chunk of 4, 256 WMMAs per wave.
// ---------------------------------------------------------------------------
__global__ void __launch_bounds__(128)
gemm_wmma_f32(const float* __restrict__ X, const float* __restrict__ W,
              float* __restrict__ Hout) {
    const int lane = threadIdx.x & 31;
    const int wave = threadIdx.x >> 5;
    const int row0 = blockIdx.x * 16;
    const int col0 = wave * 64;

    // A fragment addressing (16x4 f32): lane = M + 16*(K>=2), vgpr = K&1
    const int m  = lane & 15;
    const int k0 = (lane >> 4) << 1;   // 0 or 2
    // B fragment addressing (4x16): row striped across lanes, same half split
    const int nl = lane & 15;
    // C/D: VGPR v -> rows v (lanes 0-15) and v+8 (lanes 16-31)
    const int hs = lane >> 4;

    v8f acc[4] = {};   // zero accumulators

    const float* __restrict__ arow = X + (size_t)(row0 + m) * DIN + k0;

    #pragma unroll 4
    for (int kk = 0; kk < DIN; kk += 4) {
        v2f a = *(const v2f*)(arow + kk);                 // K = kk+k0, kk+k0+1
        const float* wrow0 = W + (size_t)(kk + k0) * HC + col0 + nl;
        const float* wrow1 = wrow0 + HC;
        #pragma unroll
        for (int t = 0; t < 4; ++t) {
            v2f b;
            b.x = wrow0[t * 16];
            b.y = wrow1[t * 16];
            acc[t] = __builtin_amdgcn_wmma_f32_16x16x4_f32(
                false, a, false, b, (short)0, acc[t], false, false);
        }
    }

    #pragma unroll
    for (int t = 0; t < 4; ++t) {
        #pragma unroll
        for (int v = 0; v < 8; ++v) {
            Hout[(size_t)(row0 + v + 8 * hs) * HC + col0 + t * 16 + nl] = acc[t][v];
        }
    }
}

// ---------------------------------------------------------------------------
// Phase 2: per-node attention logits. One wave per node; 8 channels/lane;
// lanes [8h, 8h+7] cover head h, butterfly-reduce within groups of 8.
// ---------------------------------------------------------------------------
__global__ void __launch_bounds__(256)
node_logits(const float* __restrict__ Hm, const float* __restrict__ a_src,
            const float* __restrict__ a_dst, float* __restrict__ es,
            float* __restrict__ ed) {
    int wid  = (blockIdx.x * blockDim.x + threadIdx.x) >> 5;
    int lane = threadIdx.x & 31;
    if (wid >= NNODES) return;

    const float4* hp  = (const float4*)(Hm + (size_t)wid * HC + lane * 8);
    const float4* asp = (const float4*)(a_src + lane * 8);
    const float4* adp = (const float4*)(a_dst + lane * 8);
    float4 v0 = hp[0], v1 = hp[1];
    float4 s0 = asp[0], s1 = asp[1];
    float4 d0 = adp[0], d1 = adp[1];

    float ss = v0.x * s0.x + v0.y * s0.y + v0.z * s0.z + v0.w * s0.w +
               v1.x * s1.x + v1.y * s1.y + v1.z * s1.z + v1.w * s1.w;
    float dd = v0.x * d0.x + v0.y * d0.y + v0.z * d0.z + v0.w * d0.w +
               v1.x * d1.x + v1.y * d1.y + v1.z * d1.z + v1.w * d1.w;

    #pragma unroll
    for (int off = 1; off < 8; off <<= 1) {
        ss += __shfl_xor(ss, off, 32);
        dd += __shfl_xor(dd, off, 32);
    }
    if ((lane & 7) == 0) {
        es[wid * HEADS + (lane >> 3)] = ss;
        ed[wid * HEADS + (lane >> 3)] = dd;
    }
}

// ---------------------------------------------------------------------------
// Phase 3a: segment max. Thread per edge, both directions.
// dir1: src->dst  (uses es1/ed1, max at dst)
// dir2: dst->src  (uses es2 at original dst, ed2 at original src, max at src)
// ---------------------------------------------------------------------------
__global__ void __launch_bounds__(256)
edge_max(const int* __restrict__ srcA, const int* __restrict__ dstA,
         const float* __restrict__ es1, const float* __restrict__ ed1,
         const float* __restrict__ es2, const float* __restrict__ ed2,
         float* __restrict__ emax1, float* __restrict__ emax2) {
    int t = blockIdx.x * blockDim.x + threadIdx.x;
    if (t >= ETOT) return;
    int s = (t < NEDGES) ? srcA[t] : (t - NEDGES);
    int d = (t < NEDGES) ? dstA[t] : (t - NEDGES);
    #pragma unroll
    for (int h = 0; h < HEADS; ++h) {
        float e1 = lrelu(es1[s * HEADS + h] + ed1[d * HEADS + h]);
        atomicMaxF(&emax1[d * HEADS + h], e1);
        float e2 = lrelu(es2[d * HEADS + h] + ed2[s * HEADS + h]);
        atomicMaxF(&emax2[s * HEADS + h], e2);
    }
}

// ---------------------------------------------------------------------------
// Phase 3b: numerators p = exp(e - emax[seg]) and denominators.
// ---------------------------------------------------------------------------
__global__ void __launch_bounds__(256)
edge_expsum(const int* __restrict__ srcA, const int* __restrict__ dstA,
            const float* __restrict__ es1, const float* __restrict__ ed1,
            const float* __restrict__ es2, const float* __restrict__ ed2,
            const float* __restrict__ emax1, const float* __restrict__ emax2,
            float* __restrict__ p1, float* __restrict__ p2,
            float* __restrict__ den1, float* __restrict__ den2) {
    int t = blockIdx.x * blockDim.x + threadIdx.x;
    if (t >= ETOT) return;
    int s = (t < NEDGES) ? srcA[t] : (t - NEDGES);
    int d = (t < NEDGES) ? dstA[t] : (t - NEDGES);
    #pragma unroll
    for (int h = 0; h < HEADS; ++h) {
        float e1 = lrelu(es1[s * HEADS + h] + ed1[d * HEADS + h]);
        float q1 = __expf(e1 - emax1[d * HEADS + h]);
        p1[(size_t)t * HEADS + h] = q1;
        atomicAdd(&den1[d * HEADS + h], q1);
        float e2 = lrelu(es2[d * HEADS + h] + ed2[s * HEADS + h]);
        float q2 = __expf(e2 - emax2[s * HEADS + h]);
        p2[(size_t)t * HEADS + h] = q2;
        atomicAdd(&den2[s * HEADS + h], q2);
    }
}

// ---------------------------------------------------------------------------
// Phase 4: weighted aggregation. One wave per (edge, direction); 8 channels
// per lane (lane>>3 = head). Direction weights folded into alpha so both
// directions scatter straight into d_out (L2-resident, 154 MB hot set < 192MB).
// ---------------------------------------------------------------------------
__global__ void __launch_bounds__(256)
edge_aggregate(const int* __restrict__ srcA, const int* __restrict__ dstA,
               const float* __restrict__ h1, const float* __restrict__ h2,
               const float* __restrict__ p1, const float* __restrict__ den1,
               const float* __restrict__ p2, const float* __restrict__ den2,
               float* __restrict__ out) {
    long gtid = (long)blockIdx.x * blockDim.x + threadIdx.x;
    long wid  = gtid >> 5;
    int  lane = threadIdx.x & 31;
    int  dir  = (int)(wid & 1);
    long t    = wid >> 1;
    if (t >= ETOT) return;

    int s = (t < NEDGES) ? srcA[t] : (int)(t - NEDGES);
    int d = (t < NEDGES) ? dstA[t] : (int)(t - NEDGES);
    int head = lane >> 3;
    int cb   = lane * 8;

    const float* hsrc;
    float*       outp;
    float        alpha;
    if (dir == 0) {
        alpha = (1.0f - ALPHA_W) * p1[(size_t)t * HEADS + head] /
                den1[(size_t)d * HEADS + head];
        hsrc = h1 + (size_t)s * HC;
        outp = out + (size_t)d * HC;
    } else {
        alpha = ALPHA_W * p2[(size_t)t * HEADS + head] /
                den2[(size_t)s * HEADS + head];
        hsrc = h2 + (size_t)d * HC;
        outp = out + (size_t)s * HC;
    }

    float4 v0 = *(const float4*)(hsrc + cb);
    float4 v1 = *(const float4*)(hsrc + cb + 4);
    atomicAdd(outp + cb + 0, alpha * v0.x);
    atomicAdd(outp + cb + 1, alpha * v0.y);
    atomicAdd(outp + cb + 2, alpha * v0.z);
    atomicAdd(outp + cb + 3, alpha * v0.w);
    atomicAdd(outp + cb + 4, alpha * v1.x);
    atomicAdd(outp + cb + 5, alpha * v1.y);
    atomicAdd(outp + cb + 6, alpha * v1.z);
    atomicAdd(outp + cb + 7, alpha * v1.w);
}

// ---------------------------------------------------------------------------
extern "C" void kernel_launch(void* const* d_in, const int* in_sizes, int n_in,
                              void* d_out, int out_size, void* d_ws, size_t ws_size,
                              hipStream_t stream) {
    const float* x   = (const float*)d_in[0];
    const int*   ei  = (const int*)d_in[1];
    const float* W1  = (const float*)d_in[2];
    const float* as1 = (const float*)d_in[3];
    const float* ad1 = (const float*)d_in[4];
    const float* b1  = (const float*)d_in[5];
    const float* W2  = (const float*)d_in[6];
    const float* as2 = (const float*)d_in[7];
    const float* ad2 = (const float*)d_in[8];
    const float* b2  = (const float*)d_in[9];
    const int* srcA = ei;
    const int* dstA = ei + NEDGES;
    float* out = (float*)d_out;

    // Workspace carve-up (floats).
    float* ws = (float*)d_ws;
    size_t o = 0;
    const size_t NHC = (size_t)NNODES * HC;     // 12.8M
    const size_t NH  = (size_t)NNODES * HEADS;  // 200k
    const size_t ETH = (size_t)ETOT * HEADS;    // 1.8M
    float* h1    = ws + o; o += NHC;
    float* h2    = ws + o; o += NHC;
    float* es1   = ws + o; o += NH;
    float* ed1   = ws + o; o += NH;
    float* es2   = ws + o; o += NH;
    float* ed2   = ws + o; o += NH;
    float* emax1 = ws + o; o += NH;
    float* emax2 = ws + o; o += NH;
    float* den1  = ws + o; o += NH;
    float* den2  = ws + o; o += NH;
    float* p1    = ws + o; o += ETH;
    float* p2    = ws + o; o += ETH;

    // Phase 0: init output (bias) + softmax state.
    init_kernel<<<(NNODES * HC + 255) / 256, 256, 0, stream>>>(
        out, b1, b2, emax1, emax2, den1, den2);

    // Phase 1: projections on the WMMA pipe (exact fp32).
    gemm_wmma_f32<<<NNODES / 16, 128, 0, stream>>>(x, W1, h1);
    gemm_wmma_f32<<<NNODES / 16, 128, 0, stream>>>(x, W2, h2);

    // Phase 2: per-node logits (wave per node).
    node_logits<<<(NNODES * 32 + 255) / 256, 256, 0, stream>>>(h1, as1, ad1, es1, ed1);
    node_logits<<<(NNODES * 32 + 255) / 256, 256, 0, stream>>>(h2, as2, ad2, es2, ed2);

    // Phase 3: segment softmax (max, then exp-sum).
    edge_max<<<(ETOT + 255) / 256, 256, 0, stream>>>(
        srcA, dstA, es1, ed1, es2, ed2, emax1, emax2);
    edge_expsum<<<(ETOT + 255) / 256, 256, 0, stream>>>(
        srcA, dstA, es1, ed1, es2, ed2, emax1, emax2, p1, p2, den1, den2);

    // Phase 4: weighted neighbor aggregation, both directions.
    long aggThreads = (long)ETOT * 2 * 32;
    edge_aggregate<<<(int)((aggThreads + 255) / 256), 256, 0, stream>>>(
        srcA, dstA, h1, h2, p1, den1, p2, den2, out);
}